// AVWGCN_75694503624993
// MI455X (gfx1250) — compile-verified
//
#include <hip/hip_runtime.h>
#include <hip/hip_bf16.h>

// ---------------------------------------------------------------------------
// AVWGCN on MI455X (gfx1250, wave32, WMMA).
//
// ~90 GFLOP of f32 matrix work vs ~300MB HBM traffic; all intermediates
// (S 16.8MB, T2 16.8MB, W_n 100MB, Y_k 2x33.5MB) are L2-resident (192MB L2).
// V_WMMA_F32_16X16X4_F32 preserves the f32 reference numerics.
//
//  * B panels stored TRANSPOSED in LDS -> B fragments are 2 contiguous f32
//    (single ds_load_b64 into an even VGPR pair).  LDBT=68 makes the 32-lane
//    b64 fragment reads bank-conflict-free.
//  * A panels staged with GLOBAL_LOAD_ASYNC_TO_LDS_B128 (ASYNCcnt path, no
//    VGPR round-trip).  Probe-verified prototype: param 1 is a global-AS
//    (printed '__device__') v4i*, param 2 the LDS v4i*, then offset, cpol.
//
// Fragment layouts per CDNA5 ISA 7.12.2 (wave32):
//   A 16x4 f32 (2 VGPR): lane l -> row l&15, a[v] = A[l&15][2*(l>>4)+v]
//   B 4x16 f32 (2 VGPR): lane l -> col l&15, b[v] = B[2*(l>>4)+v][l&15]
//   C/D 16x16 f32 (8 VGPR): d[v] = D[(l>>4)*8 + v][l&15]
// ---------------------------------------------------------------------------

typedef __attribute__((ext_vector_type(2))) float v2f;
typedef __attribute__((ext_vector_type(8))) float v8f;
typedef __attribute__((ext_vector_type(4))) int   v4i;

#define B_DIM   64
#define N_DIM   2048
#define C_DIM   64
#define K_CHEB  3
#define D_EMB   16

#if defined(__has_builtin)
#  if __has_builtin(__builtin_amdgcn_global_load_async_to_lds_b128)
#    define USE_ASYNC_LDS 1
#  endif
#endif

#ifdef USE_ASYNC_LDS
typedef __attribute__((address_space(1))) v4i v4i_glob;   // global AS
typedef __attribute__((address_space(3))) v4i v4i_lds;    // LDS AS
#endif

// copy 16 contiguous bytes global -> LDS
__device__ __forceinline__ void stage16(const float* g, float* l) {
#ifdef USE_ASYNC_LDS
    __builtin_amdgcn_global_load_async_to_lds_b128(
        (v4i_glob*)g, (v4i_lds*)l, 0, 0);
#else
    *(float4*)l = *(const float4*)g;
#endif
}

__device__ __forceinline__ void stage_join() {
#ifdef USE_ASYNC_LDS
#  if __has_builtin(__builtin_amdgcn_s_wait_asynccnt)
    __builtin_amdgcn_s_wait_asynccnt(0);
#  else
    asm volatile("s_wait_asynccnt 0x0" ::: "memory");
#  endif
#endif
}

// ------------------------- softmax(relu(E E^T)) ----------------------------
__launch_bounds__(256)
__global__ void relu_softmax_rows(const float* __restrict__ E,
                                  float* __restrict__ S) {
    const int CPT = N_DIM / 256;      // 8
    int n = blockIdx.x;
    int t = threadIdx.x;

    __shared__ float en[D_EMB];
    __shared__ float red[256];
    if (t < D_EMB) en[t] = E[n * D_EMB + t];
    __syncthreads();

    float z[CPT];
    float lmax = 0.0f;                // relu output >= 0
    #pragma unroll
    for (int j = 0; j < CPT; ++j) {
        int m = t + j * 256;
        const float* em = E + (size_t)m * D_EMB;
        float acc = 0.0f;
        #pragma unroll
        for (int d = 0; d < D_EMB; ++d) acc = fmaf(en[d], em[d], acc);
        acc = fmaxf(acc, 0.0f);
        z[j] = acc;
        lmax = fmaxf(lmax, acc);
    }
    red[t] = lmax; __syncthreads();
    for (int s = 128; s > 0; s >>= 1) {
        if (t < s) red[t] = fmaxf(red[t], red[t + s]);
        __syncthreads();
    }
    float rmax = red[0]; __syncthreads();

    float lsum = 0.0f;
    #pragma unroll
    for (int j = 0; j < CPT; ++j) { z[j] = __expf(z[j] - rmax); lsum += z[j]; }
    red[t] = lsum; __syncthreads();
    for (int s = 128; s > 0; s >>= 1) {
        if (t < s) red[t] += red[t + s];
        __syncthreads();
    }
    float inv = 1.0f / red[0];

    #pragma unroll
    for (int j = 0; j < CPT; ++j)
        S[(size_t)n * N_DIM + t + j * 256] = z[j] * inv;
}

// ------------------------- generic f32 WMMA GEMM ---------------------------
// C[M x Nc] = A[M x K] @ B[K x Nc] (row major), batched via blockIdx.z.
// 256 threads = 8 waves; tile 128x64; TK=64 chunks (zero padded for K<64).
// Each wave owns a 16x64 strip: one A fragment feeds 4 WMMAs per K-step.
// mode 0: C = A@B ; mode 1: C = 2*(A@B) - I  (Chebyshev T2 epilogue)
__launch_bounds__(256)
__global__ void wmma_gemm_f32(const float* __restrict__ A,
                              const float* __restrict__ B,
                              float* __restrict__ C,
                              int M, int Nc, int K,
                              size_t batchA, size_t batchB, size_t batchC,
                              int mode) {
    const int TM = 128, TN = 64, TK = 64;
    const int LDA = 68, LDBT = 68;

    __shared__ __align__(16) float Asub[TM * LDA];   // row-major [r][k]
    __shared__ __align__(16) float Bt  [TN * LDBT];  // transposed [col][k]

    A += batchA * (size_t)blockIdx.z;
    B += batchB * (size_t)blockIdx.z;
    C += batchC * (size_t)blockIdx.z;

    const int rowBase = blockIdx.x * TM;
    const int colBase = blockIdx.y * TN;
    const int t    = threadIdx.x;
    const int lane = t & 31;
    const int wave = t >> 5;          // 0..7 -> row strip
    const int lrow = lane & 15;
    const int hi   = lane >> 4;

    v8f acc[4];
    #pragma unroll
    for (int i = 0; i < 4; ++i)
        #pragma unroll
        for (int v = 0; v < 8; ++v) acc[i][v] = 0.0f;

    for (int k0 = 0; k0 < K; k0 += TK) {
        // ---- stage A chunk [TM x TK]: async global->LDS, 16B granules ----
        for (int j4 = t; j4 < (TM * TK) / 4; j4 += 256) {
            int idx = j4 * 4;
            int r = idx >> 6, c = idx & 63;
            int gc = k0 + c;
            float* dst = &Asub[r * LDA + c];
            if (gc < K)               // K is always a multiple of 4 here
                stage16(A + (size_t)(rowBase + r) * K + gc, dst);
            else
                *(float4*)dst = make_float4(0.f, 0.f, 0.f, 0.f);
        }
        // ---- stage B chunk [TK x TN], transposed into Bt[col][k] ----
        for (int j4 = t; j4 < (TK * TN) / 4; j4 += 256) {
            int idx = j4 * 4;
            int r = idx >> 6, c = idx & 63;
            int gr = k0 + r;
            float4 v = make_float4(0.f, 0.f, 0.f, 0.f);
            if (gr < K)
                v = *(const float4*)(B + (size_t)gr * Nc + colBase + c);
            Bt[(c + 0) * LDBT + r] = v.x;
            Bt[(c + 1) * LDBT + r] = v.y;
            Bt[(c + 2) * LDBT + r] = v.z;
            Bt[(c + 3) * LDBT + r] = v.w;
        }
        stage_join();
        __syncthreads();

        // prefetch next chunk into cache (global_prefetch_b8)
        if (k0 + TK < K) {
            int pr = t >> 1, pc = (t & 1) * 32;
            __builtin_prefetch(A + (size_t)(rowBase + pr) * K + (k0 + TK + pc), 0, 1);
            if (t < TK)
                __builtin_prefetch(B + (size_t)(k0 + TK + t) * Nc + colBase, 0, 1);
        }

        const float* arow = &Asub[(wave * 16 + lrow) * LDA + hi * 2];
        #pragma unroll 4
        for (int kk = 0; kk < TK; kk += 4) {
            v2f a;
            a.x = arow[kk + 0];
            a.y = arow[kk + 1];
            #pragma unroll
            for (int u = 0; u < 4; ++u) {
                const float* brow = &Bt[(u * 16 + lrow) * LDBT + hi * 2];
                v2f b;
                b.x = brow[kk + 0];
                b.y = brow[kk + 1];
                acc[u] = __builtin_amdgcn_wmma_f32_16x16x4_f32(
                    false, a, false, b, (short)0, acc[u], false, false);
            }
        }
        __syncthreads();
    }

    // ---- epilogue ----
    #pragma unroll
    for (int u = 0; u < 4; ++u) {
        int gc = colBase + u * 16 + lrow;
        #pragma unroll
        for (int v = 0; v < 8; ++v) {
            int gr = rowBase + wave * 16 + hi * 8 + v;
            float val = acc[u][v];
            if (mode == 1) val = 2.0f * val - ((gr == gc) ? 1.0f : 0.0f);
            C[(size_t)gr * Nc + gc] = val;
        }
    }
}

// --------------------- per-node contraction + bias -------------------------
// One block per node n.  out[b,n,o] = sum_{k,i} xg_k[b,n,i] * W[n,k,i,o]
//                                   + sum_d E[n,d] * bias_pool[d,o]
// A panel = [64(b) x 192(k,i)] gathered from {X, Y1, Y2}; B panel = W_n
// [192 x 64] (stored transposed per 64-chunk).  16 tiles / 8 waves.
__launch_bounds__(256)
__global__ void node_contract(const float* __restrict__ X,
                              const float* __restrict__ Y1,
                              const float* __restrict__ Y2,
                              const float* __restrict__ Wn,       // [N,3,64,64]
                              const float* __restrict__ E,        // [N,16]
                              const float* __restrict__ biasPool, // [16,64]
                              float* __restrict__ out) {
    const int LDA = 68, LDBT = 68;
    __shared__ __align__(16) float Asub[64 * LDA];
    __shared__ __align__(16) float Bt  [64 * LDBT];  // [o][i] transposed
    __shared__ float biasS[D_EMB * C_DIM];
    __shared__ float en[D_EMB];

    const int n    = blockIdx.x;
    const int t    = threadIdx.x;
    const int lane = t & 31;
    const int wave = t >> 5;
    const int lrow = lane & 15;
    const int hi   = lane >> 4;
    const int strip = wave >> 1;        // 0..3 -> b-row strip
    const int ct0   = (wave & 1) * 2;   // col tiles ct0, ct0+1

    for (int j = t; j < D_EMB * C_DIM; j += 256) biasS[j] = biasPool[j];
    if (t < D_EMB) en[t] = E[n * D_EMB + t];

    v8f acc[2];
    #pragma unroll
    for (int i = 0; i < 2; ++i)
        #pragma unroll
        for (int v = 0; v < 8; ++v) acc[i][v] = 0.0f;

    const float* srcs[K_CHEB] = {X, Y1, Y2};

    for (int kc = 0; kc < K_CHEB; ++kc) {
        __syncthreads();                 // LDS reuse + (first iter) biasS/en
        const float* src = srcs[kc];
        // A panel: Asub[b][i] = src[(b*N + n)*64 + i]  (async 16B granules)
        for (int j4 = t; j4 < (64 * 64) / 4; j4 += 256) {
            int idx = j4 * 4;
            int b = idx >> 6, i = idx & 63;
            stage16(src + ((size_t)b * N_DIM + n) * C_DIM + i,
                    &Asub[b * LDA + i]);
        }
        // B panel transposed: Bt[o][i] = Wn[n][kc][i][o]
        const float* wp = Wn + (size_t)n * (K_CHEB * C_DIM * C_DIM)
                             + (size_t)kc * (C_DIM * C_DIM);
        for (int j4 = t; j4 < (64 * 64) / 4; j4 += 256) {
            int idx = j4 * 4;
            float4 v = *(const float4*)(wp + idx);
            int r = idx >> 6, c = idx & 63;   // r = i, c = o
            Bt[(c + 0) * LDBT + r] = v.x;
            Bt[(c + 1) * LDBT + r] = v.y;
            Bt[(c + 2) * LDBT + r] = v.z;
            Bt[(c + 3) * LDBT + r] = v.w;
        }
        stage_join();
        __syncthreads();

        const float* arow = &Asub[(strip * 16 + lrow) * LDA + hi * 2];
        #pragma unroll 4
        for (int kk = 0; kk < 64; kk += 4) {
            v2f a;
            a.x = arow[kk + 0];
            a.y = arow[kk + 1];
            #pragma unroll
            for (int u = 0; u < 2; ++u) {
                const float* brow = &Bt[((ct0 + u) * 16 + lrow) * LDBT + hi * 2];
                v2f b;
                b.x = brow[kk + 0];
                b.y = brow[kk + 1];
                acc[u] = __builtin_amdgcn_wmma_f32_16x16x4_f32(
                    false, a, false, b, (short)0, acc[u], false, false);
            }
        }
    }
    __syncthreads();

    // epilogue: add bias = E[n] . bias_pool[:,o] (o constant per lane/tile)
    #pragma unroll
    for (int u = 0; u < 2; ++u) {
        int o = (ct0 + u) * 16 + lrow;
        float bias = 0.0f;
        #pragma unroll
        for (int d = 0; d < D_EMB; ++d) bias = fmaf(en[d], biasS[d * C_DIM + o], bias);
        #pragma unroll
        for (int v = 0; v < 8; ++v) {
            int b = strip * 16 + hi * 8 + v;
            out[((size_t)b * N_DIM + n) * C_DIM + o] = acc[u][v] + bias;
        }
    }
}

// ---------------------------------------------------------------------------
extern "C" void kernel_launch(void* const* d_in, const int* in_sizes, int n_in,
                              void* d_out, int out_size, void* d_ws, size_t ws_size,
                              hipStream_t stream) {
    (void)in_sizes; (void)n_in; (void)out_size; (void)ws_size;
    // inputs: [0]=graph(unused int), [1]=node_feature f32 [64,2048,64],
    // [2]=node_embeddings f32 [2048,16], [3]=weights_pool f32 [16,3,64,64],
    // [4]=bias_pool f32 [16,64]
    const float* x  = (const float*)d_in[1];
    const float* E  = (const float*)d_in[2];
    const float* Wp = (const float*)d_in[3];
    const float* bp = (const float*)d_in[4];
    float* out = (float*)d_out;

    const int N = N_DIM, Bb = B_DIM, C = C_DIM;
    const size_t NN  = (size_t)N * N;              // 4.19M floats
    const size_t BNC = (size_t)Bb * N * C;         // 8.39M floats
    const size_t WNE = (size_t)N * K_CHEB * C * C; // 25.2M floats

    float* ws = (float*)d_ws;
    float* S  = ws;            // softmax(relu(E E^T))          [N,N]
    float* T2 = S + NN;        // 2*S@S - I                     [N,N]
    float* Wn = T2 + NN;       // per-node weights              [N,3,64,64]
    float* Y1 = Wn + WNE;      // S  @ X[b]                     [B,N,C]
    float* Y2 = Y1 + BNC;      // T2 @ X[b]                     [B,N,C]

    // 1) adaptive adjacency
    relu_softmax_rows<<<N, 256, 0, stream>>>(E, S);

    // 2) Chebyshev T2 = 2*S@S - I   (2048^3 GEMM, mode 1 epilogue)
    wmma_gemm_f32<<<dim3(N / 128, N / 64, 1), 256, 0, stream>>>(
        S, S, T2, N, N, N, 0, 0, 0, 1);

    // 3) dynamic weights W_n = E @ W_pool   ([2048,16]@[16,12288])
    wmma_gemm_f32<<<dim3(N / 128, (K_CHEB * C * C) / 64, 1), 256, 0, stream>>>(
        E, Wp, Wn, N, K_CHEB * C * C, D_EMB, 0, 0, 0, 0);

    // 4/5) graph aggregation Y_k[b] = S_k @ X[b]  (batched over b)
    wmma_gemm_f32<<<dim3(N / 128, C / 64, Bb), 256, 0, stream>>>(
        S, x, Y1, N, C, N, 0, (size_t)N * C, (size_t)N * C, 0);
    wmma_gemm_f32<<<dim3(N / 128, C / 64, Bb), 256, 0, stream>>>(
        T2, x, Y2, N, C, N, 0, (size_t)N * C, (size_t)N * C, 0);

    // 6) per-node contraction + bias
    node_contract<<<N, 256, 0, stream>>>(x, Y1, Y2, Wn, E, bp, out);
}